// QuantumDeepField_32762010533983
// MI455X (gfx1250) — compile-verified
//
#include <hip/hip_runtime.h>
#include <stdint.h>

typedef __attribute__((ext_vector_type(2))) float v2f;
typedef __attribute__((ext_vector_type(8))) float v8f;
typedef __attribute__((ext_vector_type(4))) int v4i;

typedef __attribute__((address_space(1))) v4i* v4i_gptr;   // global-side pointer for async copy
typedef __attribute__((address_space(3))) v4i* v4i_lptr;   // LDS-side pointer for async copy

#define KM 128            // M (orbitals per molecule) — fixed by problem
#define TG 32             // g-rows per block tile
#define TD 128            // d-cols per block tile
#define KC 64             // m-chunk staged in LDS per iteration
#define AP 66             // padded A row length (bank-conflict free b64 loads)
#define BP 136            // padded B row length (half-wave rows map to disjoint banks; 544B = 34*16B)
#define GS_RED 16         // split-K factor for the gto-norm reduction
#define GRID15 0.16431676725154983f   // 0.3^1.5
#define EPSF 1e-12f

#if __has_builtin(__builtin_amdgcn_global_load_async_to_lds_b128)
#define USE_ASYNC_LDS 1
#else
#define USE_ASYNC_LDS 0
#endif

__device__ __forceinline__ void wait_asynccnt0() {
#if __has_builtin(__builtin_amdgcn_s_wait_asynccnt)
  __builtin_amdgcn_s_wait_asynccnt(0);
#else
  asm volatile("s_wait_asynccnt 0x0" ::: "memory");
#endif
}

// ---------------------------------------------------------------------------
// K1a: partial sums of gto^2 over the grid axis, per (b, m).
// grid: B * (M/64) * GS_RED blocks, 256 threads = 64 m-lanes x 4 g-lanes
// ---------------------------------------------------------------------------
__global__ __launch_bounds__(256)
void k_gto_norm_partial(const float* __restrict__ dmat, const int* __restrict__ qn,
                        const int* __restrict__ ao, const float* __restrict__ zeta,
                        float* __restrict__ partial, int B, int G, int M) {
  int blk = blockIdx.x;
  int gs = blk % GS_RED; blk /= GS_RED;
  int nMB = M / 64;
  int mblk = blk % nMB;
  int b = blk / nMB;
  int mi = threadIdx.x & 63;
  int gj = threadIdx.x >> 6;
  int m = mblk * 64 + mi;

  float q = (float)qn[b * M + m];
  float z = zeta[ao[b * M + m]];
  float z2 = z * z;

  int gchunk = G / GS_RED;
  int g0 = gs * gchunk;
  float s = 0.0f;
  for (int g = g0 + gj; g < g0 + gchunk; g += 4) {
    float dd = dmat[((size_t)b * G + g) * M + m];
    float p = (q < 1.5f) ? 1.0f : ((q < 2.5f) ? dd : dd * dd);   // d^(q-1), q in {1,2,3}
    float gto = p * __expf(-z2 * dd * dd);
    s += gto * gto;
  }
  __shared__ float red[256];
  red[threadIdx.x] = s;
  __syncthreads();
  if (gj == 0) {
    float t = red[mi] + red[mi + 64] + red[mi + 128] + red[mi + 192];
    partial[((size_t)b * M + m) * GS_RED + gs] = t;
  }
}

// K1b: finalize inverse denominator per (b,m)
__global__ __launch_bounds__(256)
void k_gto_norm_final(const float* __restrict__ partial, float* __restrict__ invden, int BM) {
  int i = blockIdx.x * blockDim.x + threadIdx.x;
  if (i >= BM) return;
  float s = 0.0f;
  for (int k = 0; k < GS_RED; ++k) s += partial[(size_t)i * GS_RED + k];
  float nrm = fmaxf(sqrtf(s), EPSF);
  invden[i] = 1.0f / (nrm * GRID15);
}

// ---------------------------------------------------------------------------
// K2: LCAO coefficient normalization over the orbital axis (per (b, d) column).
// ---------------------------------------------------------------------------
__global__ __launch_bounds__(256)
void k_cnorm(const float* __restrict__ coeff, const int* __restrict__ ao,
             float* __restrict__ cn, int B, int D) {
  int nDB = D / 256;
  int b = blockIdx.x / nDB;
  int dcol = (blockIdx.x % nDB) * 256 + threadIdx.x;
  __shared__ int aos[KM];
  for (int m = threadIdx.x; m < KM; m += 256) aos[m] = ao[b * KM + m];
  __syncthreads();
  float s = 0.0f;
  for (int m = 0; m < KM; ++m) {
    float v = coeff[(size_t)aos[m] * D + dcol];
    s += v * v;
  }
  float inv = 1.0f / fmaxf(sqrtf(s), EPSF);
  for (int m = 0; m < KM; ++m) {
    float v = coeff[(size_t)aos[m] * D + dcol];
    cn[((size_t)b * KM + m) * D + dcol] = v * inv;
  }
}

// ---------------------------------------------------------------------------
// K3: batched GEMM on the f32 WMMA pipe + fused per-tile mo^2 partial reduction.
//   MO[g,d]  += n_gto[g,m]  * cn[m,d]
//   LMO[g,d] += n_lgto[g,m] * cn[m,d]
// A-tiles generated from distances on the fly into LDS; B-tile staged via
// GLOBAL_LOAD_ASYNC_TO_LDS_B128 (ASYNCcnt) when available. Each block: one
// (b, 32g x 128d) tile; 8 waves x 4 f32 16x16 accumulators; K chunks of 64.
// Epilogue writes unnormalized MO/LMO and per-block sum(mo^2) partials.
// ---------------------------------------------------------------------------
__global__ __launch_bounds__(256)
void k_gemm_wmma(const float* __restrict__ dmat, const int* __restrict__ qn,
                 const int* __restrict__ ao, const float* __restrict__ zeta,
                 const float* __restrict__ invden, const float* __restrict__ cn,
                 float* __restrict__ out, float* __restrict__ psum,
                 int B, int G, int D) {
  __shared__ float Ag[TG][AP];       // normalized gto tile        [g][m]
  __shared__ float Al[TG][AP];       // normalized laplace-gto     [g][m]
  __shared__ float Bc[KC][BP];       // normalized coefficients    [m][d]
  __shared__ float red[8][32][2];    // per-(wave,lane) mo^2 partials

  int nDB = D / TD;
  int nGB = G / TG;
  int blk = blockIdx.x;
  int db = blk % nDB; blk /= nDB;
  int gb = blk % nGB;
  int b  = blk / nGB;
  int g0 = gb * TG, d0 = db * TD;

  int t = threadIdx.x;
  int lane = t & 31;
  int wave = t >> 5;               // 0..7
  int wg = wave >> 2;              // g subtile 0..1
  int wd = wave & 3;               // d subtile 0..3 (plus +4 pair)
  int row = lane & 15;
  int khalf = (lane >> 4) << 1;    // 0 or 2  (A/B fragment K split per ISA layout)
  int arow = wg * 16 + row;
  int bcol0 = wd * 16 + row;
  int bcol1 = bcol0 + 64;

  // staging decomposition for the A tile: fixed m per thread, loop over g
  int m_l = t & 63;
  int gi0 = t >> 6;                // 0..3

  v8f acc_mo0 = {}; v8f acc_mo1 = {};
  v8f acc_lm0 = {}; v8f acc_lm1 = {};

  for (int mh = 0; mh < KM; mh += KC) {
    __syncthreads();
#if USE_ASYNC_LDS
    { // stage B asynchronously: 64 x 128 chunk, 16B per lane per issue
      int dc4 = (t & 31) * 4;      // 0,4,...,124
      int m0 = t >> 5;             // 0..7
#pragma unroll
      for (int j = 0; j < KC / 8; ++j) {
        int m = m0 + 8 * j;
        const float* gsrc = cn + ((size_t)b * KM + mh + m) * D + d0 + dc4;
        __builtin_amdgcn_global_load_async_to_lds_b128(
            (v4i_gptr)(uintptr_t)gsrc,
            (v4i_lptr)(unsigned)(uintptr_t)&Bc[m][dc4], 0, 0);
      }
    }
#endif
    { // stage A: compute gto / laplace-gto for this 32 x 64 chunk
      int mg = mh + m_l;
      float q  = (float)qn[b * KM + mg];
      float z  = zeta[ao[b * KM + mg]];
      float z2 = z * z;
      float iv = invden[b * KM + mg];
#pragma unroll
      for (int j = 0; j < TG / 4; ++j) {
        int gi = gi0 + 4 * j;
        float dd = dmat[((size_t)b * G + (g0 + gi)) * KM + mg];
        float d2 = dd * dd;
        float e  = __expf(-z2 * d2) * iv;                         // fold 1/denom
        float p  = (q < 1.5f) ? 1.0f : ((q < 2.5f) ? dd : d2);    // d^(q-1)
        float poly = 4.0f * z2 * z2 * d2 * d2
                   - 2.0f * z2 * (2.0f * q + 1.0f) * d2
                   + q * (q - 1.0f);
        float gto = p * e;
        Ag[gi][m_l] = gto;
        Al[gi][m_l] = gto / d2 * poly;                            // d^(q-3) e poly / denom
      }
    }
#if USE_ASYNC_LDS
    wait_asynccnt0();
#else
    { // stage B with plain loads
      int dc = t & 127;
      int m0 = t >> 7;             // 0..1
#pragma unroll
      for (int j = 0; j < KC / 2; ++j) {
        int m = m0 + 2 * j;
        Bc[m][dc] = cn[((size_t)b * KM + mh + m) * D + d0 + dc];
      }
    }
#endif
    __syncthreads();

#pragma unroll 4
    for (int k0 = 0; k0 < KC; k0 += 4) {
      int ka = k0 + khalf;
      v2f a0, a1, b0, b1;
      a0.x = Ag[arow][ka];  a0.y = Ag[arow][ka + 1];
      a1.x = Al[arow][ka];  a1.y = Al[arow][ka + 1];
      b0.x = Bc[ka][bcol0]; b0.y = Bc[ka + 1][bcol0];
      b1.x = Bc[ka][bcol1]; b1.y = Bc[ka + 1][bcol1];
      acc_mo0 = __builtin_amdgcn_wmma_f32_16x16x4_f32(false, a0, false, b0, (short)0, acc_mo0, false, false);
      acc_mo1 = __builtin_amdgcn_wmma_f32_16x16x4_f32(false, a0, false, b1, (short)0, acc_mo1, false, false);
      acc_lm0 = __builtin_amdgcn_wmma_f32_16x16x4_f32(false, a1, false, b0, (short)0, acc_lm0, false, false);
      acc_lm1 = __builtin_amdgcn_wmma_f32_16x16x4_f32(false, a1, false, b1, (short)0, acc_lm1, false, false);
    }
  }

  // Epilogue. C/D layout: VGPR r, lanes 0-15 -> (M=r, N=lane); lanes 16-31 -> (M=r+8, N=lane-16)
  size_t half = (size_t)B * G * D;
  int n = lane & 15;
  int rbase = (lane >> 4) * 8;
  float s0 = 0.0f, s1 = 0.0f;
#pragma unroll
  for (int r = 0; r < 8; ++r) {
    size_t grow = (size_t)b * G + g0 + wg * 16 + rbase + r;
    size_t o0 = grow * D + d0 + wd * 16 + n;
    size_t o1 = o0 + 64;
    out[o0] = acc_mo0[r];
    out[o1] = acc_mo1[r];
    out[half + o0] = acc_lm0[r];
    out[half + o1] = acc_lm1[r];
    s0 += acc_mo0[r] * acc_mo0[r];
    s1 += acc_mo1[r] * acc_mo1[r];
  }
  red[wave][lane][0] = s0;
  red[wave][lane][1] = s1;
  __syncthreads();
  if (t < TD) { // one thread per tile column: combine the 4 contributing lane groups
    int j   = t >> 6;       // 0: cols 0-63 (acc0), 1: cols 64-127 (acc1)
    int c64 = t & 63;
    int wdc = c64 >> 4;
    int nn  = c64 & 15;
    float s = red[wdc][nn][j] + red[wdc][nn + 16][j]
            + red[wdc + 4][nn][j] + red[wdc + 4][nn + 16][j];
    psum[((size_t)b * D + d0 + t) * nGB + gb] = s;
  }
}

// K4: factor[b,d] = sqrt(Nel[b]/D) / max(||mo||, eps), reducing per-block partials
__global__ __launch_bounds__(256)
void k_monorm_final(const float* __restrict__ psum, const float* __restrict__ nel,
                    float* __restrict__ factor, int B, int D, int nGB) {
  int i = blockIdx.x * blockDim.x + threadIdx.x;
  if (i >= B * D) return;
  float s = 0.0f;
  for (int k = 0; k < nGB; ++k) s += psum[(size_t)i * nGB + k];
  int b = i / D;
  float nrm = fmaxf(sqrtf(s), EPSF);
  factor[i] = sqrtf(nel[b] / (float)D) / nrm;
}

// K5: in-place scale of both output halves
__global__ __launch_bounds__(256)
void k_scale(float* __restrict__ out, const float* __restrict__ factor,
             int B, int G, int D) {
  size_t half = (size_t)B * G * D;
  size_t stride = (size_t)gridDim.x * blockDim.x;
  for (size_t i = (size_t)blockIdx.x * blockDim.x + threadIdx.x; i < half; i += stride) {
    int dcol = (int)(i % D);
    int b = (int)(i / ((size_t)G * D));
    float f = factor[b * D + dcol];
    out[i] *= f;
    out[half + i] *= f;
  }
}

// ---------------------------------------------------------------------------
extern "C" void kernel_launch(void* const* d_in, const int* in_sizes, int n_in,
                              void* d_out, int out_size, void* d_ws, size_t ws_size,
                              hipStream_t stream) {
  const float* dmat  = (const float*)d_in[0];   // [B,G,M]
  const int*   qn    = (const int*)  d_in[1];   // [B,M]
  const int*   ao    = (const int*)  d_in[2];   // [B,M]
  const float* nel   = (const float*)d_in[3];   // [B]
  const float* coeff = (const float*)d_in[4];   // [N_ORB, D]
  const float* zeta  = (const float*)d_in[5];   // [N_ORB, 1]
  float* out = (float*)d_out;                   // [2, B*G, D]

  int B = in_sizes[3];
  int M = in_sizes[1] / B;                      // 128
  int G = in_sizes[0] / (B * M);                // 4096
  int NORB = in_sizes[5];
  int D = in_sizes[4] / NORB;                   // 768
  int nGB = G / TG;                             // 128

  // workspace layout (floats): ~13.6 MB total
  float* ws = (float*)d_ws;
  float* p1     = ws;                                   // B*M*GS_RED
  float* invden = p1 + (size_t)B * M * GS_RED;          // B*M
  float* cn     = invden + (size_t)B * M;               // B*M*D
  float* psum   = cn + (size_t)B * M * D;               // B*D*nGB
  float* factor = psum + (size_t)B * D * nGB;           // B*D

  k_gto_norm_partial<<<B * (M / 64) * GS_RED, 256, 0, stream>>>(dmat, qn, ao, zeta, p1, B, G, M);
  k_gto_norm_final<<<(B * M + 255) / 256, 256, 0, stream>>>(p1, invden, B * M);
  k_cnorm<<<B * (D / 256), 256, 0, stream>>>(coeff, ao, cn, B, D);
  k_gemm_wmma<<<B * (G / TG) * (D / TD), 256, 0, stream>>>(dmat, qn, ao, zeta, invden, cn, out, psum, B, G, D);
  k_monorm_final<<<(B * D + 255) / 256, 256, 0, stream>>>(psum, nel, factor, B, D, nGB);
  k_scale<<<4096, 256, 0, stream>>>(out, factor, B, G, D);
}